// PT_55576876810242
// MI455X (gfx1250) — compile-verified
//
#include <hip/hip_runtime.h>
#include <hip/hip_bf16.h>
#include <math.h>
#include <stdint.h>

// Problem constants (fixed by the reference's setup_inputs)
#define BATCH   2
#define NPTS    8192
#define KNN     16
#define NLAYERS 3
#define HPOS    32
#define HATT    12

typedef float v2f __attribute__((ext_vector_type(2)));
typedef float v4f __attribute__((ext_vector_type(4)));
typedef float v8f __attribute__((ext_vector_type(8)));

// ---------------------------------------------------------------------------
// Kernel 0: pack candidates once: pk[i] = {-2x, -2y, -2z, ||p||^2}
// ---------------------------------------------------------------------------
__global__ __launch_bounds__(256) void pack_kernel(const float* __restrict__ pos,
                                                   float* __restrict__ pk) {
    const int i = blockIdx.x * 256 + threadIdx.x;   // 0 .. B*N-1
    const float x = pos[(size_t)i * 3 + 0];
    const float y = pos[(size_t)i * 3 + 1];
    const float z = pos[(size_t)i * 3 + 2];
    v4f p;
    p.x = -2.0f * x;
    p.y = -2.0f * y;
    p.z = -2.0f * z;
    p.w = fmaf(x, x, fmaf(y, y, z * z));
    *(v4f*)&pk[(size_t)i * 4] = p;
}

// ---------------------------------------------------------------------------
// Kernel 1: exhaustive kNN on the matrix pipe, transpose-free, with
// double-buffered async global->LDS staging (ASYNCcnt path).
//
// One WMMA_F32_16X16X4_F32 computes a 16(cand) x 16(query) tile of
//     D[j][i] = ||p_j||^2 - 2 <p_i, p_j>   ( = d2(i,j) - ||p_i||^2 )
// with A[j][:] = packed candidate rows and B[:][i] = [x_i, y_i, z_i, 1].
// Queries are WMMA *columns*, so the C/D layout drops each query's distances
// straight into the two lanes (rsel, rsel+16) that own it: lane (rsel,hi)
// holds candidates hi*8+v in acc[v] -- no LDS transpose of D needed.
// The per-row constant ||p_i||^2 offset does not change per-query top-k order.
//
// Staging: supertiles of 256 packed candidates (4 KB) are copied to LDS with
// GLOBAL_LOAD_ASYNC_TO_LDS_B128 (ISA 15.18.3 op 98, GV mode, 1 x b128/lane),
// double-buffered: buffer i+1 is in flight while WMMA+selection runs on
// buffer i. Each wave has <= 2 outstanding asyncs; async loads complete in
// order, so s_wait_asynccnt 1 guarantees the older buffer is written.
//
// Selection: distances become monotonic u32 keys (IEEE radix flip) with the
// 13-bit candidate index packed in the low bits; each lane keeps a sorted
// top-16 key list via a branch-free min/max bubble (2 VALU per slot). The two
// half-lists per query are exactly merged at the end.
// ---------------------------------------------------------------------------
__global__ __launch_bounds__(256) void knn_kernel(const float* __restrict__ pos,
                                                  const float* __restrict__ pk,
                                                  int* __restrict__ idx_out) {
    const int tid  = threadIdx.x;
    const int lane = tid & 31;
    const int wave = tid >> 5;
    const int rsel = lane & 15;        // owned query (= WMMA column)
    const int hi   = lane >> 4;        // candidate half within each tile

    const int b       = blockIdx.x / (NPTS / 128);
    const int rowTile = blockIdx.x % (NPTS / 128);
    const int rowBase = rowTile * 128 + wave * 16;   // within batch
    const float* bp   = pos + (size_t)b * NPTS * 3;
    const float* bpk  = pk + (size_t)b * NPTS * 4;

    // B operand (loop-invariant query vectors):
    //  lanes 0-15 hold rows {K0=x, K1=y}; lanes 16-31 hold rows {K2=z, K3=1}
    const int qrow = rowBase + rsel;
    v2f qv;
    if (!hi) { qv.x = bp[qrow * 3 + 0]; qv.y = bp[qrow * 3 + 1]; }
    else     { qv.x = bp[qrow * 3 + 2]; qv.y = 1.0f;             }

    // Sorted (ascending) top-16 packed keys over this lane's candidate half
    unsigned bk[KNN];
#pragma unroll
    for (int t = 0; t < KNN; ++t) bk[t] = 0xFFFFFFFFu;

    // LDS: double-buffered packed candidates + merge buffer (24 KB total)
    __shared__ __attribute__((aligned(16))) float    spk[2][256 * 4];   //  8 KB
    __shared__ unsigned                              smk[8 * 32 * 16];  // 16 KB

    // Issue one async b128 per thread for a 256-candidate supertile.
    auto stage = [&](int buf, int tile0) {
        const unsigned lds_addr =
            (unsigned)(uintptr_t)&spk[buf][(size_t)tid * 4];
        const unsigned long long gaddr =
            (unsigned long long)(uintptr_t)(bpk + (size_t)(tile0 + tid) * 4);
        asm volatile("global_load_async_to_lds_b128 %0, %1, off"
                     :: "v"(lds_addr), "v"(gaddr) : "memory");
    };

    stage(0, 0);   // prologue: stage first supertile

    for (int t0 = 0; t0 < NPTS; t0 += 256) {
        const int buf = (t0 >> 8) & 1;
        if (t0 + 256 < NPTS) {
            stage(buf ^ 1, t0 + 256);                    // prefetch next tile
            asm volatile("s_wait_asynccnt 0x1" ::: "memory");
        } else {
            asm volatile("s_wait_asynccnt 0x0" ::: "memory");
        }
        __syncthreads();   // all waves' slices of the current buffer written

        const float* cur = spk[buf];
        for (int sub = 0; sub < 16; ++sub) {
            // A operand: candidate tile.
            //  lanes 0-15: {K0,K1} = {-2x,-2y}; lanes 16-31: {K2,K3} = {-2z, sq}
            const v2f am = *(const v2f*)&cur[((sub * 16 + rsel) << 2) + (hi << 1)];

            v8f acc = {};
            acc = __builtin_amdgcn_wmma_f32_16x16x4_f32(
                false, am, false, qv, (short)0, acc, false, false);

            // acc[v] = D[cand hi*8+v][query rsel] -> select directly from regs
#pragma unroll
            for (int v = 0; v < 8; ++v) {
                // monotonic key: IEEE radix flip, low 13 bits = candidate idx
                unsigned u = __float_as_uint(acc[v]);
                u ^= ((unsigned)((int)u >> 31)) | 0x80000000u;
                const unsigned key =
                    (u & 0xFFFFE000u) | (unsigned)(t0 + sub * 16 + hi * 8 + v);
                if (key < bk[KNN - 1]) {
                    // branch-free sorted insert: min/max bubble, 2 VALU/slot
                    unsigned e = key;
#pragma unroll
                    for (int t = 0; t < KNN; ++t) {
                        const unsigned mn = (bk[t] < e) ? bk[t] : e;
                        const unsigned mx = (bk[t] < e) ? e : bk[t];
                        bk[t] = mn;
                        e     = mx;
                    }
                }
            }
        }
        __syncthreads();   // everyone done with buf before it is re-staged
    }

    // Exact merge of the two per-half sorted top-16 key lists of each query.
    unsigned* mk = smk + wave * 512;
#pragma unroll
    for (int t = 0; t < KNN; ++t) mk[lane * 16 + t] = bk[t];
    // intra-wave LDS ordering; no barrier needed (each wave reads only its slice)
    if (lane < 16) {
        const unsigned* ka = mk + lane * 16;
        const unsigned* kb = mk + (lane + 16) * 16;
        int* op = idx_out + ((size_t)b * NPTS + rowBase + lane) * KNN;
        int ia = 0, ib = 0;
#pragma unroll 1
        for (int t = 0; t < KNN; ++t) {   // ia+ib = t <= 15, both stay in range
            const unsigned va = ka[ia];
            const unsigned vb = kb[ib];
            const bool ta = (va <= vb);
            op[t] = (int)((ta ? va : vb) & 8191u);
            ia += ta ? 1 : 0;
            ib += ta ? 0 : 1;
        }
    }
}

// ---------------------------------------------------------------------------
// Kernel 2: qkv = h @ w_qkv[l]  via fp32 WMMA 16x16x4 (K = D = 3, padded to 4).
// ---------------------------------------------------------------------------
__global__ __launch_bounds__(256) void qkv_kernel(const float* __restrict__ h,
                                                  const float* __restrict__ w,   // [3][9] slice
                                                  float* __restrict__ qkv) {
    const int lane = threadIdx.x & 31;
    const int wave = threadIdx.x >> 5;
    const int tile = blockIdx.x * 8 + wave;        // 16-point tile index
    const int base = tile * 16;                    // first flat point id
    const int row  = lane & 15;
    const bool hi  = lane >= 16;

    const float* hp = h + (size_t)(base + row) * 3;
    v2f a;
    if (!hi) { a.x = hp[0]; a.y = hp[1]; }
    else     { a.x = hp[2]; a.y = 0.0f;  }

    const int c = row;
    v2f bm;
    if (!hi) { bm.x = (c < 9) ? w[0 * 9 + c] : 0.0f;
               bm.y = (c < 9) ? w[1 * 9 + c] : 0.0f; }
    else     { bm.x = (c < 9) ? w[2 * 9 + c] : 0.0f;
               bm.y = 0.0f; }

    v8f acc = {};
    acc = __builtin_amdgcn_wmma_f32_16x16x4_f32(
        false, a, false, bm, (short)0, acc, false, false);

    if (c < 9) {
#pragma unroll
        for (int r = 0; r < 8; ++r) {
            const int m = r + (hi ? 8 : 0);
            qkv[(size_t)(base + m) * 9 + c] = acc[r];
        }
    }
}

// ---------------------------------------------------------------------------
// Kernel 3: fused gather + pos-MLP + attn-MLP + softmax(over K) + sigmoid.
// ---------------------------------------------------------------------------
__global__ __launch_bounds__(128) void attn_kernel(
    const float* __restrict__ qkv, const int* __restrict__ idx,
    const float* __restrict__ pos,
    const float* __restrict__ pw1, const float* __restrict__ pb1,
    const float* __restrict__ pw2, const float* __restrict__ pb2,
    const float* __restrict__ aw1, const float* __restrict__ ab1,
    const float* __restrict__ aw2, const float* __restrict__ ab2,
    float* __restrict__ hout) {
    const int tid = threadIdx.x;
    const int p   = blockIdx.x * 128 + tid;        // flat point id in [0, B*N)
    const int b   = p / NPTS;
    const int nb  = b * NPTS;

    __shared__ float s_sim[KNN * 3][128];
    __shared__ float s_ve [KNN * 3][128];
    __shared__ float s_w[HPOS * 3 + HPOS + HPOS * 3 + 3 + HATT * 3 + HATT + HATT * 3 + 3];
    float* w_pw1 = s_w;                 // 96
    float* w_pb1 = w_pw1 + HPOS * 3;    // 32
    float* w_pw2 = w_pb1 + HPOS;        // 96
    float* w_pb2 = w_pw2 + HPOS * 3;    // 3
    float* w_aw1 = w_pb2 + 3;           // 36
    float* w_ab1 = w_aw1 + HATT * 3;    // 12
    float* w_aw2 = w_ab1 + HATT;        // 36
    float* w_ab2 = w_aw2 + HATT * 3;    // 3

    if (tid < HPOS * 3) w_pw1[tid] = pw1[tid];
    if (tid < HPOS)     w_pb1[tid] = pb1[tid];
    if (tid < HPOS * 3) w_pw2[tid] = pw2[tid];
    if (tid < 3)        w_pb2[tid] = pb2[tid];
    if (tid < HATT * 3) w_aw1[tid] = aw1[tid];
    if (tid < HATT)     w_ab1[tid] = ab1[tid];
    if (tid < HATT * 3) w_aw2[tid] = aw2[tid];
    if (tid < 3)        w_ab2[tid] = ab2[tid];
    __syncthreads();

    const float q0 = qkv[(size_t)p * 9 + 0];
    const float q1 = qkv[(size_t)p * 9 + 1];
    const float q2 = qkv[(size_t)p * 9 + 2];
    const float px = pos[(size_t)p * 3 + 0];
    const float py = pos[(size_t)p * 3 + 1];
    const float pz = pos[(size_t)p * 3 + 2];

    for (int j = 0; j < KNN; ++j) {
        const int n = nb + idx[(size_t)p * KNN + j];
        const float kx = qkv[(size_t)n * 9 + 3], ky = qkv[(size_t)n * 9 + 4], kz = qkv[(size_t)n * 9 + 5];
        const float vx = qkv[(size_t)n * 9 + 6], vy = qkv[(size_t)n * 9 + 7], vz = qkv[(size_t)n * 9 + 8];
        const float rx = px - pos[(size_t)n * 3 + 0];
        const float ry = py - pos[(size_t)n * 3 + 1];
        const float rz = pz - pos[(size_t)n * 3 + 2];

        // rel_emb = relu(rel_pos @ pw1 + pb1) @ pw2 + pb2
        float re0 = w_pb2[0], re1 = w_pb2[1], re2 = w_pb2[2];
        for (int h = 0; h < HPOS; ++h) {
            float t = fmaf(rx, w_pw1[h], fmaf(ry, w_pw1[HPOS + h], fmaf(rz, w_pw1[2 * HPOS + h], w_pb1[h])));
            t = fmaxf(t, 0.0f);
            re0 = fmaf(t, w_pw2[h * 3 + 0], re0);
            re1 = fmaf(t, w_pw2[h * 3 + 1], re1);
            re2 = fmaf(t, w_pw2[h * 3 + 2], re2);
        }

        // sim = relu((q - k_j + rel_emb) @ aw1 + ab1) @ aw2 + ab2
        const float x0 = q0 - kx + re0;
        const float x1 = q1 - ky + re1;
        const float x2 = q2 - kz + re2;
        float s0 = w_ab2[0], s1 = w_ab2[1], s2 = w_ab2[2];
        for (int h = 0; h < HATT; ++h) {
            float t = fmaf(x0, w_aw1[h], fmaf(x1, w_aw1[HATT + h], fmaf(x2, w_aw1[2 * HATT + h], w_ab1[h])));
            t = fmaxf(t, 0.0f);
            s0 = fmaf(t, w_aw2[h * 3 + 0], s0);
            s1 = fmaf(t, w_aw2[h * 3 + 1], s1);
            s2 = fmaf(t, w_aw2[h * 3 + 2], s2);
        }
        s_sim[j * 3 + 0][tid] = s0;
        s_sim[j * 3 + 1][tid] = s1;
        s_sim[j * 3 + 2][tid] = s2;
        s_ve [j * 3 + 0][tid] = vx + re0;
        s_ve [j * 3 + 1][tid] = vy + re1;
        s_ve [j * 3 + 2][tid] = vz + re2;
    }

    // Per-dim softmax over neighbors, weighted sum, sigmoid
    float hn[3];
    for (int d = 0; d < 3; ++d) {
        float m = s_sim[d][tid];
        for (int j = 1; j < KNN; ++j) m = fmaxf(m, s_sim[j * 3 + d][tid]);
        float se = 0.0f, ag = 0.0f;
        for (int j = 0; j < KNN; ++j) {
            const float e = __expf(s_sim[j * 3 + d][tid] - m);
            se += e;
            ag = fmaf(e, s_ve[j * 3 + d][tid], ag);
        }
        ag /= se;
        hn[d] = 1.0f / (1.0f + __expf(-ag));
    }
    hout[(size_t)p * 3 + 0] = hn[0];
    hout[(size_t)p * 3 + 1] = hn[1];
    hout[(size_t)p * 3 + 2] = hn[2];
}

// ---------------------------------------------------------------------------
// Kernel 4: final linear (3 -> 2) + softmax over the 2 outputs
// ---------------------------------------------------------------------------
__global__ __launch_bounds__(256) void head_kernel(const float* __restrict__ h,
                                                   const float* __restrict__ lw,
                                                   const float* __restrict__ lb,
                                                   float* __restrict__ out) {
    const int p = blockIdx.x * 256 + threadIdx.x;
    const float h0 = h[(size_t)p * 3 + 0];
    const float h1 = h[(size_t)p * 3 + 1];
    const float h2 = h[(size_t)p * 3 + 2];
    const float z0 = lb[0] + h0 * lw[0] + h1 * lw[2] + h2 * lw[4];
    const float z1 = lb[1] + h0 * lw[1] + h1 * lw[3] + h2 * lw[5];
    const float m  = fmaxf(z0, z1);
    const float e0 = __expf(z0 - m);
    const float e1 = __expf(z1 - m);
    const float inv = 1.0f / (e0 + e1);
    out[(size_t)p * 2 + 0] = e0 * inv;
    out[(size_t)p * 2 + 1] = e1 * inv;
}

// ---------------------------------------------------------------------------
extern "C" void kernel_launch(void* const* d_in, const int* in_sizes, int n_in,
                              void* d_out, int out_size, void* d_ws, size_t ws_size,
                              hipStream_t stream) {
    const float* feats = (const float*)d_in[0];
    const float* pos   = (const float*)d_in[1];
    // d_in[2] = mask (unused by the reference forward pass)
    const float* w_qkv = (const float*)d_in[3];
    const float* pw1   = (const float*)d_in[4];
    const float* pb1   = (const float*)d_in[5];
    const float* pw2   = (const float*)d_in[6];
    const float* pb2   = (const float*)d_in[7];
    const float* aw1   = (const float*)d_in[8];
    const float* ab1   = (const float*)d_in[9];
    const float* aw2   = (const float*)d_in[10];
    const float* ab2   = (const float*)d_in[11];
    const float* lw    = (const float*)d_in[12];
    const float* lb    = (const float*)d_in[13];
    float* out = (float*)d_out;

    // Workspace layout (~2 MB total); pk first for 16B alignment.
    char* ws = (char*)d_ws;
    float* pkbuf = (float*)ws;                                        // B*N*4 floats (256 KB)
    int*   idx   = (int*)(ws + (size_t)BATCH * NPTS * 4 * sizeof(float));
    float* qkv   = (float*)((char*)idx + (size_t)BATCH * NPTS * KNN * sizeof(int));
    float* hbuf  = qkv + (size_t)BATCH * NPTS * 9;

    pack_kernel<<<(BATCH * NPTS) / 256, 256, 0, stream>>>(pos, pkbuf);
    knn_kernel<<<BATCH * (NPTS / 128), 256, 0, stream>>>(pos, pkbuf, idx);

    const float* hin = feats;
    for (int l = 0; l < NLAYERS; ++l) {
        qkv_kernel<<<(BATCH * NPTS) / (16 * 8), 256, 0, stream>>>(hin, w_qkv + l * 27, qkv);
        attn_kernel<<<(BATCH * NPTS) / 128, 128, 0, stream>>>(
            qkv, idx, pos,
            pw1 + l * (HPOS * 3), pb1 + l * HPOS, pw2 + l * (HPOS * 3), pb2 + l * 3,
            aw1 + l * (HATT * 3), ab1 + l * HATT, aw2 + l * (HATT * 3), ab2 + l * 3,
            hbuf);
        hin = hbuf;
    }
    head_kernel<<<(BATCH * NPTS) / 256, 256, 0, stream>>>(hbuf, lw, lb, out);
}